// LogNorm_74113955659815
// MI455X (gfx1250) — compile-verified
//
#include <hip/hip_runtime.h>
#include <hip/hip_bf16.h>

// LogNorm moment-matching for MI455X (gfx1250, wave32, WMMA fp32 path).
//
// S=16 states, C=64 cell types, G=20000 genes.
// var[s,g] = a^T corr a with a[c] = cell_prob[s,c]*exp(mu+0.5 s^2)*sigma.
// Flattened: M = corr(64x64) @ A(64 x S*G) -> 2.6 GFLOP; I/O ~12.8 MB.
// fp32 WMMA 16x16x4 keeps reference precision; HBM traffic is unchanged by
// precision so there is nothing to gain from f16/bf16 here.

#define S_STATES 16
#define C_TYPES  64
#define G_GENES  20000
#define GB       16   // genes per workgroup

typedef __attribute__((ext_vector_type(2))) float v2f;
typedef __attribute__((ext_vector_type(8))) float v8f;

__global__ __launch_bounds__(128)
void lognorm_wmma_kernel(const float* __restrict__ Z_mu,
                         const float* __restrict__ Z_sigma,
                         const float* __restrict__ corr,
                         const float* __restrict__ cell_prob,
                         float* __restrict__ mu_out,
                         float* __restrict__ sg_out) {
    __shared__ float s_corr[C_TYPES * C_TYPES]; // 16 KB
    __shared__ float s_ev[C_TYPES * GB];        // exp(mu + 0.5*sigma^2)
    __shared__ float s_ab[C_TYPES * GB];        // ev * sigma
    __shared__ float s_cp[S_STATES * C_TYPES];  // cell_prob

    const int tid  = threadIdx.x;
    const int g0   = blockIdx.x * GB;

    // Stage corr (4096 f32) and cell_prob (1024 f32) into LDS.
    #pragma unroll
    for (int i = tid; i < C_TYPES * C_TYPES; i += 128) s_corr[i] = corr[i];
    #pragma unroll
    for (int i = tid; i < S_STATES * C_TYPES; i += 128) s_cp[i] = cell_prob[i];

    // One exp per (c, gene) for this block, shared by all 16 states.
    #pragma unroll
    for (int i = tid; i < C_TYPES * GB; i += 128) {
        const int c = i >> 4;          // 0..63
        const int g = i & 15;          // 0..15
        const float m  = Z_mu[c * G_GENES + g0 + g];
        const float sg = Z_sigma[c * G_GENES + g0 + g];
        const float e  = __expf(m + 0.5f * sg * sg);
        s_ev[i] = e;
        s_ab[i] = e * sg;
    }
    __syncthreads();

    const int wave  = tid >> 5;        // 0..3
    const int lane  = tid & 31;
    const int m     = lane & 15;       // row (A-op) / gene column (B,C,D)
    const int hi    = lane >> 4;       // half-wave select
    const int khalf = hi << 1;         // K base: 0 or 2

    // Hoist corr A-operands: 4 row-tiles x 16 K-chunks, 2 VGPRs each (128 VGPRs),
    // reused across 4 state iterations.
    v2f cA[4][16];
    #pragma unroll
    for (int rt = 0; rt < 4; ++rt) {
        #pragma unroll
        for (int kc = 0; kc < 16; ++kc) {
            const int row = rt * 16 + m;
            const int d0  = kc * 4 + khalf;
            cA[rt][kc].x = s_corr[row * 64 + d0];
            cA[rt][kc].y = s_corr[row * 64 + d0 + 1];
        }
    }

    // Each wave handles states {wave, wave+4, wave+8, wave+12}.
    #pragma unroll
    for (int si = 0; si < 4; ++si) {
        const int s = wave + si * 4;
        const float* cp = &s_cp[s * C_TYPES];

        // M = corr @ A_s : 64x16 output as 4 WMMA D-tiles, K=64 in 16 chunks.
        v8f acc0 = {}, acc1 = {}, acc2 = {}, acc3 = {};
        #pragma unroll
        for (int kc = 0; kc < 16; ++kc) {
            const int c0 = kc * 4 + khalf;
            v2f b;
            b.x = cp[c0]     * s_ab[c0 * GB + m];
            b.y = cp[c0 + 1] * s_ab[(c0 + 1) * GB + m];
            acc0 = __builtin_amdgcn_wmma_f32_16x16x4_f32(false, cA[0][kc], false, b, (short)0, acc0, false, false);
            acc1 = __builtin_amdgcn_wmma_f32_16x16x4_f32(false, cA[1][kc], false, b, (short)0, acc1, false, false);
            acc2 = __builtin_amdgcn_wmma_f32_16x16x4_f32(false, cA[2][kc], false, b, (short)0, acc2, false, false);
            acc3 = __builtin_amdgcn_wmma_f32_16x16x4_f32(false, cA[3][kc], false, b, (short)0, acc3, false, false);
        }

        // Ssum[g]: half-waves split c-range, combine with shfl_xor(16).
        float ss = 0.0f;
        const int cbase = hi * 32;
        #pragma unroll
        for (int c = 0; c < 32; ++c)
            ss += cp[cbase + c] * s_ev[(cbase + c) * GB + m];
        ss += __shfl_xor(ss, 16, 32);

        // Diagonal contraction: var_num = sum_c A_s[c,g] * M[c,g].
        // D-tile layout: lane covers row mm = rt*16 + v + 8*hi, col g = m.
        float dot = 0.0f;
        #pragma unroll
        for (int v = 0; v < 8; ++v) {
            const int r0 = v + hi * 8;
            dot += acc0[v] * (cp[r0]      * s_ab[(r0)      * GB + m]);
            dot += acc1[v] * (cp[r0 + 16] * s_ab[(r0 + 16) * GB + m]);
            dot += acc2[v] * (cp[r0 + 32] * s_ab[(r0 + 32) * GB + m]);
            dot += acc3[v] * (cp[r0 + 48] * s_ab[(r0 + 48) * GB + m]);
        }
        dot += __shfl_xor(dot, 16, 32);

        const float var   = dot / (ss * ss);
        const float mu_o  = __logf(ss) - 0.5f * var;
        const float sig_o = sqrtf(var);

        if (lane < 16) {
            mu_out[s * G_GENES + g0 + m] = mu_o;
            sg_out[s * G_GENES + g0 + m] = sig_o;
        }
    }
}

extern "C" void kernel_launch(void* const* d_in, const int* in_sizes, int n_in,
                              void* d_out, int out_size, void* d_ws, size_t ws_size,
                              hipStream_t stream) {
    const float* Z_mu      = (const float*)d_in[0];
    const float* Z_sigma   = (const float*)d_in[1];
    const float* corr      = (const float*)d_in[2];
    const float* cell_prob = (const float*)d_in[3];

    float* mu_out = (float*)d_out;                      // (S, G)
    float* sg_out = mu_out + S_STATES * G_GENES;        // (S, G)

    const int nblocks = G_GENES / GB;                   // 1250, exact
    lognorm_wmma_kernel<<<nblocks, 128, 0, stream>>>(
        Z_mu, Z_sigma, corr, cell_prob, mu_out, sg_out);
}